// LIF_30949534335421
// MI455X (gfx1250) — compile-verified
//
#include <hip/hip_runtime.h>

// ---------------------------------------------------------------------------
// LIF Euler scan, forward value:
//   v = v - v/20 + I[t];  s = (v>=1);  out[t]=s;  v = s ? 0 : v
// [B=131072, L=400] f32 in, f32 out.  Pure streaming: ~420MB @ 23.3TB/s.
// Strategy: stage time-chunks through LDS with CDNA5 async global<->LDS DMA
// (ASYNCcnt), double buffered; per-thread sequential integration of one row.
// ---------------------------------------------------------------------------

#define LSTEPS   400
#define CHUNK    8          // timesteps per staged chunk (32B per row)
#define NCHUNK   (LSTEPS / CHUNK)   // 50
#define TPB      256        // 8 waves (wave32)
#define STRIDE   12         // floats per row in LDS tile (pad; 48B, 16B-aligned)

// Types exactly as clang-22 declares the async-DMA builtins:
//   param0: int __vector(4) __device__ *   (v4i in AS1/global)
//   param1: v4i in AS3/shared (expected)
typedef int v4i __attribute__((__vector_size__(16)));
typedef v4i __attribute__((address_space(1)))* gv4_p;   // global v4i*
typedef v4i __attribute__((address_space(3)))* lv4_p;   // LDS v4i*

// --- split dependency-counter waits (CDNA5) --------------------------------
// Raw asm with "memory" clobbers so the compiler cannot reorder the LDS
// tile accesses across the waits.
#define S_WAIT_ASYNCCNT(n) asm volatile("s_wait_asynccnt " #n ::: "memory")
#define S_WAIT_DSCNT(n)    asm volatile("s_wait_dscnt "    #n ::: "memory")

// --- async DMA: global -> LDS (16B per lane), tracked by ASYNCcnt ----------
static __device__ __forceinline__ void async_load16(const float* g, float* l) {
#if __has_builtin(__builtin_amdgcn_global_load_async_to_lds_b128)
    __builtin_amdgcn_global_load_async_to_lds_b128(
        (gv4_p)(void*)(const_cast<float*>(g)), (lv4_p)(void*)l, 0, 0);
#else
    unsigned loff = (unsigned)(unsigned long long)(lv4_p)(void*)l;
    asm volatile("global_load_async_to_lds_b128 %0, %1, off"
                 :: "v"(loff), "v"(g) : "memory");
#endif
}

// --- async DMA: LDS -> global (16B per lane), tracked by ASYNCcnt ----------
static __device__ __forceinline__ void async_store16(float* g, float* l) {
#if __has_builtin(__builtin_amdgcn_global_store_async_from_lds_b128)
    __builtin_amdgcn_global_store_async_from_lds_b128(
        (gv4_p)(void*)g, (lv4_p)(void*)l, 0, 0);
#else
    unsigned loff = (unsigned)(unsigned long long)(lv4_p)(void*)l;
    asm volatile("global_store_async_from_lds_b128 %0, %1, off"
                 :: "v"(g), "v"(loff) : "memory");
#endif
}

// one LIF Euler step; emits spike, applies reset
static __device__ __forceinline__ float lif_step(float& v, float It) {
    v = v - v * (1.0f / 20.0f) + It;
    float s = (v >= 1.0f) ? 1.0f : 0.0f;
    v = (v >= 1.0f) ? 0.0f : v;
    return s;
}

__global__ __launch_bounds__(TPB)
void lif_scan_kernel(const float* __restrict__ in, float* __restrict__ out) {
    __shared__ __align__(16) float s_in [2][TPB * STRIDE];
    __shared__ __align__(16) float s_out[2][TPB * STRIDE];

    const int tid = threadIdx.x;
    const long long row = (long long)blockIdx.x * TPB + tid;
    const float* gin  = in  + row * LSTEPS;
    float*       gout = out + row * LSTEPS;

    float* lin [2] = { &s_in [0][tid * STRIDE], &s_in [1][tid * STRIDE] };
    float* lout[2] = { &s_out[0][tid * STRIDE], &s_out[1][tid * STRIDE] };

    // prologue: prefetch chunk 0 (2 x b128 = 8 timesteps)
    async_load16(gin + 0, lin[0] + 0);
    async_load16(gin + 4, lin[0] + 4);

    float v = 0.0f;

    for (int c = 0; c < NCHUNK; ++c) {
        const int buf = c & 1;

        // stream in chunk c+1 while we integrate chunk c
        if (c + 1 < NCHUNK) {
            const float* gsrc = gin + (c + 1) * CHUNK;
            float* ldst = lin[buf ^ 1];
            async_load16(gsrc + 0, ldst + 0);
            async_load16(gsrc + 4, ldst + 4);
            // ASYNCcnt <= 2: loads complete in order, so chunk-c's loads
            // (older than the 2 just issued) are done; stores complete in
            // order among stores, so the async stores that read lout[buf]
            // two iterations ago are also done before we overwrite it.
            S_WAIT_ASYNCCNT(2);
        } else {
            S_WAIT_ASYNCCNT(0);
        }

        // integrate 8 timesteps from the LDS tile
#pragma unroll
        for (int k = 0; k < 2; ++k) {
            float4 x = *(const float4*)(lin[buf] + 4 * k);
            float4 r;
            r.x = lif_step(v, x.x);
            r.y = lif_step(v, x.y);
            r.z = lif_step(v, x.z);
            r.w = lif_step(v, x.w);
            *(float4*)(lout[buf] + 4 * k) = r;
        }

        // make the DS writes visible to the async-DMA engine, then push the
        // spike chunk out LDS -> HBM (write-combined 128b ops)
        S_WAIT_DSCNT(0);
        float* gdst = gout + c * CHUNK;
        async_store16(gdst + 0, lout[buf] + 0);
        async_store16(gdst + 4, lout[buf] + 4);
    }
    // s_endpgm performs an implicit wait-idle, draining remaining async stores
}

// fallback for a ragged tail (rows % TPB != 0) — not hit for B=131072
__global__ void lif_tail_kernel(const float* __restrict__ in,
                                float* __restrict__ out,
                                long long row0, long long rows) {
    long long row = row0 + blockIdx.x * blockDim.x + threadIdx.x;
    if (row >= rows) return;
    const float* gin  = in  + row * LSTEPS;
    float*       gout = out + row * LSTEPS;
    float v = 0.0f;
    for (int t = 0; t < LSTEPS; ++t) {
        gout[t] = lif_step(v, gin[t]);
    }
}

extern "C" void kernel_launch(void* const* d_in, const int* in_sizes, int n_in,
                              void* d_out, int out_size, void* d_ws, size_t ws_size,
                              hipStream_t stream) {
    (void)n_in; (void)out_size; (void)d_ws; (void)ws_size;
    const float* I = (const float*)d_in[0];
    float* out = (float*)d_out;

    const long long total = in_sizes[0];          // B * L
    const long long rows  = total / LSTEPS;       // 131072
    const long long full  = rows / TPB;           // 512 full blocks
    const long long tail  = rows - full * TPB;    // 0 for this problem

    if (full > 0) {
        lif_scan_kernel<<<(unsigned)full, TPB, 0, stream>>>(I, out);
    }
    if (tail > 0) {
        unsigned tb = (unsigned)((tail + 255) / 256);
        lif_tail_kernel<<<tb, 256, 0, stream>>>(I, out, full * TPB, rows);
    }
}